// NodeEdgeFeatureEnhancer_35287451304770
// MI455X (gfx1250) — compile-verified
//
#include <hip/hip_runtime.h>
#include <hip/hip_bf16.h>

// CDNA5 / gfx1250: wave32, WMMA 16x16x4 fp32 path.
typedef __attribute__((ext_vector_type(2))) float v2f;
typedef __attribute__((ext_vector_type(8))) float v8f;

#define IN_DIM 32        // input dim of both MLP layers
#define LDS_STRIDE 36    // 16 rows * 36 floats; 36*r mod 64 hits 16 distinct banks, keeps 8B alignment
#define WAVES_PER_BLOCK 8

__device__ __forceinline__ v8f wmma4(v2f a, v2f b, v8f c) {
  // D = A(16x4,f32) * B(4x16,f32) + C(16x16,f32)
  return __builtin_amdgcn_wmma_f32_16x16x4_f32(
      /*neg_a=*/false, a, /*neg_b=*/false, b,
      /*c_mod=*/(short)0, c, /*reuse_a=*/false, /*reuse_b=*/false);
}

// Single-instruction ReLU: median(x, 0, +inf) == max(x, 0) for finite x.
__device__ __forceinline__ float relu1(float x) {
  return __builtin_amdgcn_fmed3f(x, 0.0f, __builtin_inff());
}

// B fragment (4x16 slice of W^T) for k-block kb, n-tile nt.
// y = x @ W^T  =>  B[k][n] = W[n][k], W row-major [32 x 32].
// Lane layout (mirrors A): lane = n%16 + 16*(k_in_block>=2), vgpr = k_in_block%2.
__device__ __forceinline__ v2f load_bfrag(const float* __restrict__ W,
                                          int lane, int kb, int nt) {
  int n = nt * 16 + (lane & 15);
  int k = kb * 4 + ((lane >> 4) << 1);   // even -> 8B aligned
  return *(const v2f*)(W + n * IN_DIM + k);
}

struct Frags {
  v2f b1[2][8];   // layer-1 weights, [n_tile][k_block]
  v2f b2[2][8];   // layer-2 weights
  float bias1[2];
  float bias2[2];
};

__device__ __forceinline__ void load_frags(Frags& f,
                                           const float* __restrict__ w1,
                                           const float* __restrict__ b1,
                                           const float* __restrict__ w2,
                                           const float* __restrict__ b2,
                                           int lane) {
#pragma unroll
  for (int nt = 0; nt < 2; ++nt) {
#pragma unroll
    for (int kb = 0; kb < 8; ++kb) {
      f.b1[nt][kb] = load_bfrag(w1, lane, kb, nt);
      f.b2[nt][kb] = load_bfrag(w2, lane, kb, nt);
    }
    f.bias1[nt] = b1[nt * 16 + (lane & 15)];
    f.bias2[nt] = b2[nt * 16 + (lane & 15)];
  }
}

// Two-layer ReLU MLP on a 16-row tile. Output y (16x32) returned as two
// 16x16 C/D fragments: lane L, reg r holds y[r + 8*(L>>4)][nt*16 + (L&15)].
// X is streamed exactly once -> non-temporal loads keep MI455X's 192MB L2
// free for the atomic scatter-max region and the weights.
__device__ __forceinline__ void mlp_tile(const Frags& f,
                                         const float* __restrict__ X,
                                         int row_base, int n_rows,
                                         float* __restrict__ wlds,
                                         int lane, v8f& y0, v8f& y1) {
  const int m = lane & 15;
  const int half = lane >> 4;
  int mrow = row_base + ((m < n_rows) ? m : (n_rows - 1));  // clamp partial tiles
  const v2f* xrow = (const v2f*)(X + (size_t)mrow * IN_DIM + half * 2);

  v8f acc0, acc1;
#pragma unroll
  for (int r = 0; r < 8; ++r) { acc0[r] = f.bias1[0]; acc1[r] = f.bias1[1]; }
#pragma unroll
  for (int kb = 0; kb < 8; ++kb) {
    // A frag: K = kb*4 + 2*half + {0,1}; NT: streamed, bypass far-cache retention
    v2f a = __builtin_nontemporal_load(xrow + kb * 2);
    acc0 = wmma4(a, f.b1[0][kb], acc0);
    acc1 = wmma4(a, f.b1[1][kb], acc1);
  }
#pragma unroll
  for (int r = 0; r < 8; ++r) { acc0[r] = relu1(acc0[r]); acc1[r] = relu1(acc1[r]); }

  // h -> per-wave LDS (row-major, padded stride), cross-lane transpose to A layout
  {
    float* base = wlds + (8 * half) * LDS_STRIDE + m;   // col = lane&15
#pragma unroll
    for (int r = 0; r < 8; ++r) {
      base[r * LDS_STRIDE]      = acc0[r];
      base[r * LDS_STRIDE + 16] = acc1[r];
    }
  }
  __builtin_amdgcn_wave_barrier();
  asm volatile("s_wait_dscnt 0x0" ::: "memory");

  // layer 2: A frags from LDS
#pragma unroll
  for (int r = 0; r < 8; ++r) { acc0[r] = f.bias2[0]; acc1[r] = f.bias2[1]; }
  const float* hrow = wlds + m * LDS_STRIDE + half * 2;
#pragma unroll
  for (int kb = 0; kb < 8; ++kb) {
    v2f a = *(const v2f*)(hrow + kb * 4);
    acc0 = wmma4(a, f.b2[0][kb], acc0);
    acc1 = wmma4(a, f.b2[1][kb], acc1);
  }
  __builtin_amdgcn_wave_barrier();           // keep next iter's LDS stores after these loads
  asm volatile("" ::: "memory");
#pragma unroll
  for (int r = 0; r < 8; ++r) { y0[r] = relu1(acc0[r]); y1[r] = relu1(acc1[r]); }
}

__global__ void __launch_bounds__(256) zero_out_kernel(float* __restrict__ out, int n) {
  int i = blockIdx.x * blockDim.x + threadIdx.x;
  int stride = gridDim.x * blockDim.x;
  float4* o4 = (float4*)out;
  int n4 = n >> 2;  // n = n_nodes*64, divisible by 4
  float4 z; z.x = 0.f; z.y = 0.f; z.z = 0.f; z.w = 0.f;
  // regular (RT) stores: the agg half of this region is about to be hammered
  // by ~51M atomics; keep the lines resident in L2.
  for (; i < n4; i += stride) o4[i] = z;
}

__global__ void __launch_bounds__(256) node_mlp_kernel(
    const float* __restrict__ X,
    const float* __restrict__ w1, const float* __restrict__ b1,
    const float* __restrict__ w2, const float* __restrict__ b2,
    float* __restrict__ out, int n_nodes) {
  __shared__ float lds[WAVES_PER_BLOCK * 16 * LDS_STRIDE];
  const int lane = threadIdx.x & 31;
  const int wid  = threadIdx.x >> 5;
  float* wlds = lds + wid * 16 * LDS_STRIDE;

  Frags f;
  load_frags(f, w1, b1, w2, b2, lane);

  const int ntiles = (n_nodes + 15) >> 4;
  const int gwave  = blockIdx.x * WAVES_PER_BLOCK + wid;
  const int nwaves = gridDim.x * WAVES_PER_BLOCK;
  const int half = lane >> 4, col = lane & 15;

  for (int tile = gwave; tile < ntiles; tile += nwaves) {
    const int row_base = tile * 16;
    int nr = n_nodes - row_base; if (nr > 16) nr = 16;
    v8f y0, y1;
    mlp_tile(f, X, row_base, nr, wlds, lane, y0, y1);
#pragma unroll
    for (int r = 0; r < 8; ++r) {
      int row = row_base + r + 8 * half;
      if (row < n_nodes) {
        float* o = out + (size_t)row * 64;    // node_emb -> cols [0,32)
        // written once, never re-read by this dispatch -> non-temporal
        __builtin_nontemporal_store(y0[r], o + col);
        __builtin_nontemporal_store(y1[r], o + 16 + col);
      }
    }
  }
}

__global__ void __launch_bounds__(256) edge_mlp_scatter_kernel(
    const float* __restrict__ X, const int* __restrict__ src,
    const float* __restrict__ w1, const float* __restrict__ b1,
    const float* __restrict__ w2, const float* __restrict__ b2,
    float* __restrict__ out, int n_edges) {
  __shared__ float lds[WAVES_PER_BLOCK * 16 * LDS_STRIDE];
  const int lane = threadIdx.x & 31;
  const int wid  = threadIdx.x >> 5;
  float* wlds = lds + wid * 16 * LDS_STRIDE;

  Frags f;
  load_frags(f, w1, b1, w2, b2, lane);

  const int ntiles = (n_edges + 15) >> 4;
  const int gwave  = blockIdx.x * WAVES_PER_BLOCK + wid;
  const int nwaves = gridDim.x * WAVES_PER_BLOCK;
  const int half = lane >> 4, col = lane & 15;

  for (int tile = gwave; tile < ntiles; tile += nwaves) {
    const int row_base = tile * 16;
    int nr = n_edges - row_base; if (nr > 16) nr = 16;

    // prefetch next tile's rows into cache hierarchy
    int ntile = tile + nwaves;
    if (ntile < ntiles) {
      const float* p = X + (size_t)(ntile * 16 + (lane & 15)) * IN_DIM;
      __builtin_prefetch(p, 0, 1);
    }

    v8f y0, y1;
    mlp_tile(f, X, row_base, nr, wlds, lane, y0, y1);

    // scatter-max into out[src][32:64). Post-ReLU values are >= 0, so IEEE
    // float ordering == unsigned-bit ordering; zero-init == reference's
    // "empty segment -> 0" semantics. Default (RT) atomics: target region
    // (12.8 MB) stays resident in the 192 MB L2.
#pragma unroll
    for (int r = 0; r < 8; ++r) {
      int row = row_base + r + 8 * half;
      if (row < n_edges) {
        int s = src[row];
        unsigned* dst = (unsigned*)(out + (size_t)s * 64 + 32);
        atomicMax(dst + col,      __float_as_uint(y0[r]));
        atomicMax(dst + 16 + col, __float_as_uint(y1[r]));
      }
    }
  }
}

extern "C" void kernel_launch(void* const* d_in, const int* in_sizes, int n_in,
                              void* d_out, int out_size, void* d_ws, size_t ws_size,
                              hipStream_t stream) {
  const float* node_features = (const float*)d_in[0];
  const float* edge_features = (const float*)d_in[1];
  const int*   edge_index    = (const int*)d_in[2];   // [2, E] flat; first E = src
  const float* nw1 = (const float*)d_in[3];
  const float* nb1 = (const float*)d_in[4];
  const float* nw2 = (const float*)d_in[5];
  const float* nb2 = (const float*)d_in[6];
  const float* ew1 = (const float*)d_in[7];
  const float* eb1 = (const float*)d_in[8];
  const float* ew2 = (const float*)d_in[9];
  const float* eb2 = (const float*)d_in[10];
  float* out = (float*)d_out;

  const int n_nodes = in_sizes[0] / 32;
  const int n_edges = in_sizes[1] / 32;

  zero_out_kernel<<<512, 256, 0, stream>>>(out, n_nodes * 64);
  node_mlp_kernel<<<512, 256, 0, stream>>>(
      node_features, nw1, nb1, nw2, nb2, out, n_nodes);
  edge_mlp_scatter_kernel<<<2048, 256, 0, stream>>>(
      edge_features, edge_index, ew1, eb1, ew2, eb2, out, n_edges);
}